// Net_27882927686181
// MI455X (gfx1250) — compile-verified
//
#include <hip/hip_runtime.h>
#include <cstdint>
#include <cstddef>

namespace {
constexpr int BN = 1024;   // batch
constexpr int MN = 2048;   // nodes per layer
constexpr int KN = 32;     // fan-in
constexpr int LN = 12;     // layers
constexpr int MT = 16;     // m-tile per block
constexpr int BT = 256;    // b-tile per block
}

#define AS1 __attribute__((address_space(1)))
#define AS3 __attribute__((address_space(3)))

typedef int v4i_ __attribute__((vector_size(16)));

#if __has_builtin(__builtin_amdgcn_global_load_async_to_lds_b128) && \
    __has_builtin(__builtin_amdgcn_s_wait_asynccnt)
#define USE_ASYNC_LDS 1
#else
#define USE_ASYNC_LDS 0
#endif

__device__ __forceinline__ float sigmoidf_(float t) {
    return 1.0f / (1.0f + __expf(-t));
}

// Force wave-uniform values into SGPRs (legal: used only where value is uniform per wave).
__device__ __forceinline__ int bcast_i(int v) {
#if defined(__HIP_DEVICE_COMPILE__)
    return __builtin_amdgcn_readfirstlane(v);
#else
    return v;
#endif
}
__device__ __forceinline__ float bcast_f(float v) {
#if defined(__HIP_DEVICE_COMPILE__)
    return __uint_as_float((unsigned)__builtin_amdgcn_readfirstlane((int)__float_as_uint(v)));
#else
    return v;
#endif
}

// One layer: voutT[m, b] = sigmoid(sum_k W[m,k] * vinT[idx[m,k], b] + bias[m])
// Block: MT=16 nodes x BT=256 batch columns. 256 threads = 4 groups of 64 (2 waves);
// group g handles nodes g, g+4, g+8, g+12; each thread owns 4 batch columns (float4).
// Within a wave, (m,k) are uniform -> idx/W broadcast to SGPRs, gather base is scalar.
__global__ __launch_bounds__(256) void dag_layer(const float* __restrict__ vinT,
                                                 float* __restrict__ voutT,
                                                 const int* __restrict__ idx_l,
                                                 const float* __restrict__ w_l,
                                                 const float* __restrict__ bias_l) {
    __shared__ int   s_idx[MT * KN];
    __shared__ float s_w[MT * KN];
    __shared__ float s_b[MT];

    const int t      = threadIdx.x;
    const int m0     = blockIdx.x * MT;
    const int bbase  = blockIdx.y * BT + (t & 63) * 4;
    const int mgroup = t >> 6;

    const int*   gidx = idx_l + (size_t)m0 * KN;
    const float* gw   = w_l + (size_t)m0 * KN;

#if USE_ASYNC_LDS
    // CDNA5 async global->LDS staging of this tile's idx/W (2KB + 2KB).
    if (t < 128) {
        __builtin_amdgcn_global_load_async_to_lds_b128(
            (AS1 v4i_*)(gidx + t * 4), (AS3 v4i_*)(s_idx + t * 4), 0, 0);
    } else {
        const int u = t - 128;
        __builtin_amdgcn_global_load_async_to_lds_b128(
            (AS1 v4i_*)(gw + u * 4), (AS3 v4i_*)(s_w + u * 4), 0, 0);
    }
    if (t < MT) s_b[t] = bias_l[m0 + t];
    __builtin_amdgcn_s_wait_asynccnt(0);   // each wave waits for its own async ops
    __syncthreads();                       // then all waves' LDS writes are visible
#else
    if (t < 128) {
        reinterpret_cast<int4*>(s_idx)[t] = reinterpret_cast<const int4*>(gidx)[t];
    } else {
        reinterpret_cast<float4*>(s_w)[t - 128] = reinterpret_cast<const float4*>(gw)[t - 128];
    }
    if (t < MT) s_b[t] = bias_l[m0 + t];
    __syncthreads();
#endif

#pragma unroll
    for (int mi = 0; mi < MT / 4; ++mi) {
        const int m = mgroup + mi * 4;
        const float bb = s_b[m];
        float4 acc = make_float4(bb, bb, bb, bb);
#pragma unroll 8
        for (int k = 0; k < KN; ++k) {
            // wave-uniform: lift into SGPRs so the gather base is scalar (SALU address math)
            const int   j = bcast_i(s_idx[m * KN + k]);
            const float w = bcast_f(s_w[m * KN + k]);
            const float* row = vinT + (size_t)j * BN;   // scalar base
            // contiguous across batch: 64 lanes x 16B = 1KB segment (L2-resident)
            const float4 g = *reinterpret_cast<const float4*>(row + bbase);
            acc.x = fmaf(w, g.x, acc.x);
            acc.y = fmaf(w, g.y, acc.y);
            acc.z = fmaf(w, g.z, acc.z);
            acc.w = fmaf(w, g.w, acc.w);
        }
        float4 r;
        r.x = sigmoidf_(acc.x);
        r.y = sigmoidf_(acc.y);
        r.z = sigmoidf_(acc.z);
        r.w = sigmoidf_(acc.w);
        *reinterpret_cast<float4*>(voutT + (size_t)(m0 + m) * BN + bbase) = r;
    }
}

// v0T[m, b] = sigmoid(x[b, m] + 1), fused transpose via LDS tile.
__global__ __launch_bounds__(256) void dag_init_transpose(const float* __restrict__ x,
                                                          float* __restrict__ vT) {
    __shared__ float tile[32][33];
    const int m0   = blockIdx.x * 32;
    const int b0   = blockIdx.y * 32;
    const int lane = threadIdx.x & 31;
    const int row  = threadIdx.x >> 5;
#pragma unroll
    for (int i = 0; i < 4; ++i) {
        const int bl = row + i * 8;
        const float t = x[(size_t)(b0 + bl) * MN + (m0 + lane)] + 1.0f;  // coalesced over m
        tile[bl][lane] = sigmoidf_(t);                                    // tile[b_local][m_local]
    }
    __syncthreads();
#pragma unroll
    for (int i = 0; i < 4; ++i) {
        const int ml = row + i * 8;
        vT[(size_t)(m0 + ml) * BN + (b0 + lane)] = tile[lane][ml];        // coalesced over b
    }
}

// out[b, m] = vT[m, b]
__global__ __launch_bounds__(256) void dag_final_transpose(const float* __restrict__ vT,
                                                           float* __restrict__ out) {
    __shared__ float tile[32][33];
    const int b0   = blockIdx.x * 32;
    const int m0   = blockIdx.y * 32;
    const int lane = threadIdx.x & 31;
    const int row  = threadIdx.x >> 5;
#pragma unroll
    for (int i = 0; i < 4; ++i) {
        const int ml = row + i * 8;
        tile[ml][lane] = vT[(size_t)(m0 + ml) * BN + (b0 + lane)];        // coalesced over b
    }
    __syncthreads();
#pragma unroll
    for (int i = 0; i < 4; ++i) {
        const int bl = row + i * 8;
        out[(size_t)(b0 + bl) * MN + (m0 + lane)] = tile[lane][bl];       // coalesced over m
    }
}

extern "C" void kernel_launch(void* const* d_in, const int* in_sizes, int n_in,
                              void* d_out, int out_size, void* d_ws, size_t ws_size,
                              hipStream_t stream) {
    const float* x    = (const float*)d_in[0];
    const int*   idx  = (const int*)d_in[1];
    const float* W    = (const float*)d_in[2];
    const float* bias = (const float*)d_in[3];
    float* out = (float*)d_out;

    float* vA = (float*)d_ws;  // transposed values [MN, BN], 8 MB
    float* vB = out;           // d_out reused as the other ping-pong buffer (same 8 MB shape)

    dim3 blk(256);
    dag_init_transpose<<<dim3(MN / 32, BN / 32), blk, 0, stream>>>(x, vA);

    for (int l = 0; l < LN; ++l) {
        const float* src = (l & 1) ? vB : vA;
        float*       dst = (l & 1) ? vA : vB;
        dag_layer<<<dim3(MN / MT, BN / BT), blk, 0, stream>>>(
            src, dst,
            idx + (size_t)l * MN * KN,
            W   + (size_t)l * MN * KN,
            bias + (size_t)l * MN);
    }
    // After 12 (even) layers the transposed result is in vA; emit [B, M] into d_out.
    dag_final_transpose<<<dim3(BN / 32, MN / 32), blk, 0, stream>>>(vA, out);
}